// Encoder_76931454206576
// MI455X (gfx1250) — compile-verified
//
#include <hip/hip_runtime.h>
#include <cmath>

// Problem constants from the reference.
#define BATCH 1024
#define SEQ   512
#define DIM   768

typedef float v2f __attribute__((ext_vector_type(2)));
typedef float v8f __attribute__((ext_vector_type(8)));

// ---------------------------------------------------------------------------
// Kernel 1: masked sum pooling  [B,S,D] -> [B,D]
// HBM-bound: 1.6 GB of embeddings. One block per batch row; 192 threads each
// own one float4 column slice (192*4 = 768). Mask staged in LDS; mask is
// uniform across the block so the skip-branch is wave-coherent and halves
// HBM traffic on average (random 0/1 mask).
// ---------------------------------------------------------------------------
__global__ __launch_bounds__(192) void pool_mask_kernel(
    const float* __restrict__ x,      // [B,S,D]
    const int*   __restrict__ mask,   // [B,S]
    float*       __restrict__ pooled) // [B,D]
{
    const int b = blockIdx.x;
    __shared__ int smask[SEQ];
    for (int s = threadIdx.x; s < SEQ; s += 192)
        smask[s] = mask[(size_t)b * SEQ + s];
    __syncthreads();

    float4 acc = make_float4(0.f, 0.f, 0.f, 0.f);
    const float4* row = (const float4*)(x + (size_t)b * SEQ * DIM) + threadIdx.x;
    const int stride4 = DIM / 4; // 192 float4 per sequence row

    #pragma unroll 4
    for (int s = 0; s < SEQ; ++s) {
        if (smask[s]) {                       // uniform per block -> scalar branch
            float4 v = row[(size_t)s * stride4]; // global_load_b128, coalesced
            acc.x += v.x; acc.y += v.y; acc.z += v.z; acc.w += v.w;
        }
    }
    ((float4*)(pooled + (size_t)b * DIM))[threadIdx.x] = acc;
}

// ---------------------------------------------------------------------------
// Kernels 2/3: fused  out = tanh(A @ W + bias)  using V_WMMA_F32_16X16X4_F32.
// A: [M,K] row-major, W: [K,N] row-major, out: [M,N]. M=1024, K=N=768.
//
// Each wave computes a 32x32 C tile = 2x2 grid of 16x16 WMMA tiles.
// K loop advances by 4 (the f32 WMMA K depth): per step load 2 A fragments
// (contiguous float2 per lane) + 2 B fragments (2 coalesced dwords per lane),
// then issue 4 v_wmma_f32_16x16x4_f32 with f32 accumulation.
//
// Fragment layouts (ISA 7.12.2, wave32):
//   A 16x4 : lane l<16 -> row=l,    holds K = k+0,k+1 ; lane l>=16 -> K = k+2,k+3
//   B 4x16 : VGPR0 = rows {k+0 | k+2}, VGPR1 = rows {k+1 | k+3}, col = lane&15
//   C 16x16: VGPR j, lane l -> row = j + (l>>4)*8, col = lane&15
// ---------------------------------------------------------------------------
__global__ __launch_bounds__(256) void wmma_gemm_bias_tanh_kernel(
    const float* __restrict__ A,     // [M,K]
    const float* __restrict__ W,     // [K,N]
    const float* __restrict__ bias,  // [N]
    float*       __restrict__ out,   // [M,N]
    int M, int K, int N)
{
    const int wave  = threadIdx.x >> 5;
    const int lane  = threadIdx.x & 31;
    const int tilesN = N / 32;                       // 24
    const int tile   = blockIdx.x * 8 + wave;        // 768 wave-tiles total
    const int m0 = (tile / tilesN) * 32;
    const int n0 = (tile % tilesN) * 32;

    const int lrow  = lane & 15;       // row within 16-row half / col within tile
    const int khalf = (lane >> 4) * 2; // 0 for lanes 0-15, 2 for lanes 16-31

    v8f c00 = {}; v8f c01 = {}; v8f c10 = {}; v8f c11 = {};

    // A fragment base pointers: element [m0 + lrow][khalf], contiguous pair.
    const float* a0p = A + (size_t)(m0 + lrow)      * K + khalf;
    const float* a1p = A + (size_t)(m0 + 16 + lrow) * K + khalf;
    // B fragment base pointers: element [khalf][n0 + lrow]; .y is one row down.
    const float* b0p = W + (size_t)khalf * N + (n0 + lrow);
    const float* b1p = W + (size_t)khalf * N + (n0 + 16 + lrow);

    for (int k = 0; k < K; k += 4) {
        v2f a0 = *(const v2f*)(a0p + k);         // global_load_b64
        v2f a1 = *(const v2f*)(a1p + k);
        v2f b0, b1;
        const size_t koff = (size_t)k * N;
        b0[0] = b0p[koff];     b0[1] = b0p[koff + N];
        b1[0] = b1p[koff];     b1[1] = b1p[koff + N];

        // D = A x B + C, f32 in / f32 accumulate (matches reference numerics).
        c00 = __builtin_amdgcn_wmma_f32_16x16x4_f32(false, a0, false, b0,
                                                    (short)0, c00, false, false);
        c01 = __builtin_amdgcn_wmma_f32_16x16x4_f32(false, a0, false, b1,
                                                    (short)0, c01, false, false);
        c10 = __builtin_amdgcn_wmma_f32_16x16x4_f32(false, a1, false, b0,
                                                    (short)0, c10, false, false);
        c11 = __builtin_amdgcn_wmma_f32_16x16x4_f32(false, a1, false, b1,
                                                    (short)0, c11, false, false);
    }

    // Epilogue: bias + tanh, then store. C layout per ISA 7.12.2.
    const int colA = n0 + lrow;        // N sub-tile 0
    const int colB = n0 + 16 + lrow;   // N sub-tile 1
    const float biasA = bias[colA];
    const float biasB = bias[colB];
    const int rbase0 = m0 + ((lane >> 4) * 8);        // M sub-tile 0
    const int rbase1 = m0 + 16 + ((lane >> 4) * 8);   // M sub-tile 1

    #pragma unroll
    for (int j = 0; j < 8; ++j) {
        const size_t r0 = (size_t)(rbase0 + j) * N;
        const size_t r1 = (size_t)(rbase1 + j) * N;
        out[r0 + colA] = tanhf(c00[j] + biasA);
        out[r0 + colB] = tanhf(c01[j] + biasB);
        out[r1 + colA] = tanhf(c10[j] + biasA);
        out[r1 + colB] = tanhf(c11[j] + biasB);
    }
}

// ---------------------------------------------------------------------------
// Host-side launcher. Inputs (setup_inputs order):
//   0: token_embeds f32 [B,S,D]   1: attention_mask int [B,S]
//   2: W1 f32 [D,D]  3: b1 f32 [D]  4: W2 f32 [D,D]  5: b2 f32 [D]
// Output: f32 [B,D]. Workspace: pooled (3 MB) + h (3 MB).
// ---------------------------------------------------------------------------
extern "C" void kernel_launch(void* const* d_in, const int* in_sizes, int n_in,
                              void* d_out, int out_size, void* d_ws, size_t ws_size,
                              hipStream_t stream)
{
    const float* token_embeds = (const float*)d_in[0];
    const int*   attn_mask    = (const int*)  d_in[1];
    const float* W1           = (const float*)d_in[2];
    const float* b1           = (const float*)d_in[3];
    const float* W2           = (const float*)d_in[4];
    const float* b2           = (const float*)d_in[5];
    float*       out          = (float*)d_out;

    float* pooled = (float*)d_ws;                       // [B,D]
    float* h      = pooled + (size_t)BATCH * DIM;       // [B,D]

    // 1) masked sum pooling (HBM-bound)
    pool_mask_kernel<<<BATCH, 192, 0, stream>>>(token_embeds, attn_mask, pooled);

    // 2) h = tanh(pooled @ W1 + b1)
    const int tiles  = (BATCH / 32) * (DIM / 32);       // 768 wave-tiles
    const int blocks = tiles / 8;                       // 8 waves per 256-thd block
    wmma_gemm_bias_tanh_kernel<<<blocks, 256, 0, stream>>>(
        pooled, W1, b1, h, BATCH, DIM, DIM);

    // 3) out = tanh(h @ W2 + b2)
    wmma_gemm_bias_tanh_kernel<<<blocks, 256, 0, stream>>>(
        h, W2, b2, out, BATCH, DIM, DIM);
}